// lossFunc_53609781789270
// MI455X (gfx1250) — compile-verified
//
#include <hip/hip_runtime.h>
#include <hip/hip_bf16.h>

// Problem constants (from reference): B=512, T=200, NUM_Q=2048, NUM_S=512
#define B_DIM  512
#define T_DIM  200
#define T1     199
#define NUM_Q_ 2048
#define NUM_S_ 512

typedef float v2f __attribute__((ext_vector_type(2)));
typedef float v8f __attribute__((ext_vector_type(8)));

// Numerically stable softplus: log(1 + exp(x)) = max(x,0) + log1p(exp(-|x|))
__device__ __forceinline__ float softplus_f(float x) {
    return fmaxf(x, 0.0f) + log1pf(expf(-fabsf(x)));
}

// One block per batch row b. 256 threads (8 wave32s); threads t < 199 are "real".
// Phase 1: int-max tree reduce for `last` (mask extent).
// Phase 2: gather xs/xq, BCE via stable softplus, write masked pred/gt,
//          float-sum tree reduce, thread 0 emits per_student[b] = sum / cnt.
__global__ void loss_main_kernel(const float* __restrict__ p_s,
                                 const float* __restrict__ p_q,
                                 const int*   __restrict__ batch,
                                 float* __restrict__ pred_out,
                                 float* __restrict__ gt_out,
                                 float* __restrict__ per_student) {
    __shared__ int   s_last[256];
    __shared__ float s_sum[256];

    const int t = threadIdx.x;
    const int b = blockIdx.x;

    int q_id = 0, s_id = 0, ans = 0;
    int cand = -1;
    if (t < T1) {
        const int* bp = batch + ((size_t)b * T_DIM + (t + 1)) * 3;
        q_id = bp[0];
        s_id = bp[1];
        ans  = bp[2];
        cand = (s_id > 0) ? t : -1;
    }
    s_last[t] = cand;
    __syncthreads();
    #pragma unroll
    for (int off = 128; off > 0; off >>= 1) {
        if (t < off) s_last[t] = max(s_last[t], s_last[t + off]);
        __syncthreads();
    }
    const int last = s_last[0];
    __syncthreads();

    float local = 0.0f;
    if (t < T1) {
        const float xs = p_s[((size_t)b * T1 + t) * NUM_S_ + (size_t)(s_id - 1)];
        const float xq = p_q[((size_t)b * T1 + t) * NUM_Q_ + (size_t)(q_id - 1)];
        const float a  = (float)ans;

        const float ps = 1.0f / (1.0f + expf(-xs));
        const float pq = 1.0f / (1.0f + expf(-xq));
        const float p  = 0.5f * (ps + pq);

        const float mask = (t <= last) ? 1.0f : 0.0f;

        pred_out[(size_t)b * T1 + t] = p * mask;
        gt_out  [(size_t)b * T1 + t] = a * mask;

        // -log_sigmoid(x) = softplus(-x); -log_sigmoid(-x) = softplus(x)
        const float bce_s = a * softplus_f(-xs) + (1.0f - a) * softplus_f(xs);
        const float bce_q = a * softplus_f(-xq) + (1.0f - a) * softplus_f(xq);
        local = mask * (bce_s + bce_q);
    }
    s_sum[t] = local;
    __syncthreads();
    #pragma unroll
    for (int off = 128; off > 0; off >>= 1) {
        if (t < off) s_sum[t] += s_sum[t + off];
        __syncthreads();
    }
    if (t == 0) {
        per_student[b] = s_sum[0] / (float)(last + 1);
    }
}

// Final 512-element sum on the WMMA pipe: 8 chained V_WMMA_F32_16X16X4_F32.
// A = 64 partials per step (16x4 f32 tile: 2 VGPRs x 32 lanes), B = ones(4x16)
// (all-ones is layout-invariant), C/D chained accumulator.
// D[m][n] = sum over the 8 steps of row-sums -> every distinct input lands in
// exactly one (m,k) slot, so sum over column N=0 = total. C/D layout: VGPR r,
// lane 0  -> (M=r,   N=0); lane 16 -> (M=r+8, N=0).
// Single wave of 32 threads -> EXEC all ones as WMMA requires. Deterministic.
__global__ void loss_reduce_wmma(const float* __restrict__ per_student,
                                 float* __restrict__ out_loss) {
    __shared__ float lds[32 * 8];
    const int lane = threadIdx.x;  // 0..31

    v8f acc = {};
    const v2f ones = {1.0f, 1.0f};

    #pragma unroll
    for (int i = 0; i < 8; ++i) {
        v2f a;
        a.x = per_student[i * 64 + lane * 2 + 0];
        a.y = per_student[i * 64 + lane * 2 + 1];
        acc = __builtin_amdgcn_wmma_f32_16x16x4_f32(
            /*neg_a=*/false, a, /*neg_b=*/false, ones,
            /*c_mod=*/(short)0, acc, /*reuse_a=*/false, /*reuse_b=*/false);
    }

    #pragma unroll
    for (int r = 0; r < 8; ++r) lds[lane * 8 + r] = acc[r];
    __syncthreads();

    if (lane == 0) {
        float total = 0.0f;
        #pragma unroll
        for (int r = 0; r < 8; ++r) total += lds[0 * 8 + r] + lds[16 * 8 + r];
        out_loss[0] = total;
    }
}

extern "C" void kernel_launch(void* const* d_in, const int* in_sizes, int n_in,
                              void* d_out, int out_size, void* d_ws, size_t ws_size,
                              hipStream_t stream) {
    const float* p_s   = (const float*)d_in[0];   // (B, T1, NUM_S) f32
    const float* p_q   = (const float*)d_in[1];   // (B, T1, NUM_Q) f32
    const int*   batch = (const int*)  d_in[2];   // (B, T, 3) integer

    float* out  = (float*)d_out;
    float* loss = out;                                  // [0]
    float* pred = out + 1;                              // [1, 1+B*T1)
    float* gt   = out + 1 + (size_t)B_DIM * T1;         // [1+B*T1, 1+2*B*T1)

    float* per_student = (float*)d_ws;                  // 512 floats of scratch

    loss_main_kernel<<<B_DIM, 256, 0, stream>>>(p_s, p_q, batch, pred, gt, per_student);
    loss_reduce_wmma<<<1, 32, 0, stream>>>(per_student, loss);
}